// ForwardXRayVolumeRenderer_26207890440562
// MI455X (gfx1250) — compile-verified
//
#include <hip/hip_runtime.h>
#include <math.h>

typedef float v2f __attribute__((ext_vector_type(2)));
typedef float v8f __attribute__((ext_vector_type(8)));

#define IMG  256
#define NPTS 320

// ---------------------------------------------------------------------------
// Kernel 1: ray-march render. One block per image row h, one thread per column w.
// Writes screen value directly into transposed output position out[w*IMG + h].
// Per-wave: 2x V_WMMA_F32_16X16X4_F32 compute rdir = dir @ R^T for 32 rays,
// redistributed to per-ray lanes through LDS.
// ---------------------------------------------------------------------------
__global__ __launch_bounds__(256) void render_kernel(
    const float* __restrict__ vol,   // (256,256,256) z-major
    const float* __restrict__ Rm,    // 3x3 row-major
    const float* __restrict__ Tv,    // 3
    const float* __restrict__ fo,    // 1
    float* __restrict__ out)         // 256*256, transposed layout
{
  __shared__ float s_rdir[256][3];

  const int t    = threadIdx.x;
  const int h    = blockIdx.x;
  const int w    = t;
  const int lane = t & 31;
  const int wq   = t >> 5;

  const float f = fo[0];
  const float R00=Rm[0],R01=Rm[1],R02=Rm[2];
  const float R10=Rm[3],R11=Rm[4],R12=Rm[5];
  const float R20=Rm[6],R21=Rm[7],R22=Rm[8];

#if __has_builtin(__builtin_amdgcn_wmma_f32_16x16x4_f32)
  {
    // B (4x16): B[k][n] = R^T[k][n] = R[n][k]; layout mirrors 16x4 A:
    //   VGPR0: lanes0-15 row K=0, lanes16-31 row K=2; VGPR1: K=1 / K=3.
    const int n  = lane & 15;
    const int kb = (lane >> 4) * 2;           // 0 or 2
    v2f Bm; Bm.x = 0.f; Bm.y = 0.f;
    if (n < 3) {
      Bm.x = (kb     < 3) ? Rm[n*3 + kb    ] : 0.f;
      Bm.y = (kb + 1 < 3) ? Rm[n*3 + kb + 1] : 0.f;
    }
    const float diry = (255.f/256.f - (float)h * (1.f/128.f)) / f;
    #pragma unroll
    for (int half = 0; half < 2; ++half) {
      // A (16x4): row m = ray (wq*32 + half*16 + m); A = [dirx, diry, 1, 0].
      // Layout: lanes0-15 hold K=0,1 of row M=lane; lanes16-31 hold K=2,3.
      const int   m    = lane & 15;
      const int   wm   = wq*32 + half*16 + m;
      const float dirx = (255.f/256.f - (float)wm * (1.f/128.f)) / f;
      v2f Am;
      Am.x = (lane < 16) ? dirx : 1.0f;   // K=0 : K=2 (dirz == 1)
      Am.y = (lane < 16) ? diry : 0.0f;   // K=1 : K=3 (pad)
      v8f Dm = {};
      Dm = __builtin_amdgcn_wmma_f32_16x16x4_f32(
               false, Am, false, Bm, (short)0, Dm, false, false);
      // D[m][n]: VGPR v holds rows M=v (lanes0-15) and M=v+8 (lanes16-31); N=lane%16.
      if (n < 3) {
        #pragma unroll
        for (int v = 0; v < 8; ++v) {
          const int mr = v + 8 * (lane >> 4);
          s_rdir[wq*32 + half*16 + mr][n] = Dm[v];
        }
      }
    }
  }
#else
  {
    const float dirx = (255.f/256.f - (float)w * (1.f/128.f)) / f;
    const float diry = (255.f/256.f - (float)h * (1.f/128.f)) / f;
    s_rdir[t][0] = dirx*R00 + diry*R01 + R02;
    s_rdir[t][1] = dirx*R10 + diry*R11 + R12;
    s_rdir[t][2] = dirx*R20 + diry*R21 + R22;
  }
#endif
  __syncthreads();

  const float rdx = s_rdir[t][0];
  const float rdy = s_rdir[t][1];
  const float rdz = s_rdir[t][2];

  // origin = -(T @ R^T); voxel-space: p = d*(rdir*128) + (origin*128 + 127.5)
  const float Tx=Tv[0], Ty=Tv[1], Tz=Tv[2];
  const float ox = -(Tx*R00 + Ty*R01 + Tz*R02);
  const float oy = -(Tx*R10 + Ty*R11 + Tz*R12);
  const float oz = -(Tx*R20 + Ty*R21 + Tz*R22);
  const float rvx = rdx*128.f, rvy = rdy*128.f, rvz = rdz*128.f;
  const float bvx = ox*128.f + 127.5f;
  const float bvy = oy*128.f + 127.5f;
  const float bvz = oz*128.f + 127.5f;

  float Ttr = 1.f;   // running prod(1 - dens)  (EPS=1e-10 is a no-op in f32)
  float feat = 0.f;

  for (int i = 0; i < NPTS; ++i) {
    const float d  = fmaf((float)i, 6.f/319.f, 3.f);
    const float px = fmaf(d, rvx, bvx);
    const float py = fmaf(d, rvy, bvy);
    const float pz = fmaf(d, rvz, bvz);

    // Early-out: no corner can contribute => dens = 0, transmittance unchanged.
    if (px > -1.f && px < 256.f && py > -1.f && py < 256.f &&
        pz > -1.f && pz < 256.f) {
      const float xf = floorf(px), yf = floorf(py), zf = floorf(pz);
      const float fx = px - xf, fy = py - yf, fz = pz - zf;
      const int x0 = (int)xf, y0 = (int)yf, z0 = (int)zf;

      // Per-axis masked weights (validity factorizes per axis).
      const float wx0 = (x0 >= 0 ) ? (1.f - fx) : 0.f;
      const float wx1 = (x0 <= 254) ? fx        : 0.f;
      const float wy0 = (y0 >= 0 ) ? (1.f - fy) : 0.f;
      const float wy1 = (y0 <= 254) ? fy        : 0.f;
      const float wz0 = (z0 >= 0 ) ? (1.f - fz) : 0.f;
      const float wz1 = (z0 <= 254) ? fz        : 0.f;

      const int cx0 = x0 < 0 ? 0 : x0;  const int cx1 = x0 >= 255 ? 255 : x0 + 1;
      const int cy0 = y0 < 0 ? 0 : y0;  const int cy1 = y0 >= 255 ? 255 : y0 + 1;
      const int cz0 = z0 < 0 ? 0 : z0;  const int cz1 = z0 >= 255 ? 255 : z0 + 1;

      const int zo0 = cz0 << 16, zo1 = cz1 << 16;
      const int yo0 = cy0 << 8,  yo1 = cy1 << 8;

      const float v000 = vol[zo0 + yo0 + cx0], v001 = vol[zo0 + yo0 + cx1];
      const float v010 = vol[zo0 + yo1 + cx0], v011 = vol[zo0 + yo1 + cx1];
      const float v100 = vol[zo1 + yo0 + cx0], v101 = vol[zo1 + yo0 + cx1];
      const float v110 = vol[zo1 + yo1 + cx0], v111 = vol[zo1 + yo1 + cx1];

      const float gray =
          wz0 * (wy0 * (wx0*v000 + wx1*v001) + wy1 * (wx0*v010 + wx1*v011)) +
          wz1 * (wy0 * (wx0*v100 + wx1*v101) + wy1 * (wx0*v110 + wx1*v111));
      const float wsum = (wx0 + wx1) * (wy0 + wy1) * (wz0 + wz1);

      const float dens = 0.1f * wsum;
      feat = fmaf(dens * Ttr, gray, feat);   // absorption BEFORE update
      Ttr *= (1.f - dens);
    }
  }

  // screen = (3*feat + opacity)/4, opacity = 1 - prod(1-dens) = 1 - Ttr
  out[w * IMG + h] = (3.f * feat + (1.f - Ttr)) * 0.25f;
}

// ---------------------------------------------------------------------------
// Kernel 2: single-block reduction -> {mean, std(ddof=1), min, max}
// ---------------------------------------------------------------------------
__global__ __launch_bounds__(256) void reduce_kernel(
    const float* __restrict__ x, float* __restrict__ stats)
{
  __shared__ double ss[256], sq[256];
  __shared__ float  mn[256], mx[256];
  const int t = threadIdx.x;
  double a = 0.0, b = 0.0;
  float lo = 3.4e38f, hi = -3.4e38f;
  for (int i = t; i < IMG * IMG; i += 256) {
    const float v = x[i];
    a += (double)v;
    b += (double)v * (double)v;
    lo = fminf(lo, v);
    hi = fmaxf(hi, v);
  }
  ss[t] = a; sq[t] = b; mn[t] = lo; mx[t] = hi;
  __syncthreads();
  for (int s = 128; s > 0; s >>= 1) {
    if (t < s) {
      ss[t] += ss[t + s];
      sq[t] += sq[t + s];
      mn[t] = fminf(mn[t], mn[t + s]);
      mx[t] = fmaxf(mx[t], mx[t + s]);
    }
    __syncthreads();
  }
  if (t == 0) {
    const double N = (double)(IMG * IMG);
    const double mean = ss[0] / N;
    double var = (sq[0] - N * mean * mean) / (N - 1.0);
    if (var < 0.0) var = 0.0;
    stats[0] = (float)mean;
    stats[1] = (float)sqrt(var);
    stats[2] = mn[0];
    stats[3] = mx[0];
  }
}

// ---------------------------------------------------------------------------
// Kernel 3: fused z-score + min-max normalize, in place.
// out = ((x - min)/(sd+1e-8) + 1e-8) / ((max - min)/(sd+1e-8) + 1e-8)
// ---------------------------------------------------------------------------
__global__ __launch_bounds__(256) void norm_kernel(
    float* __restrict__ x, const float* __restrict__ stats)
{
  const int i = blockIdx.x * 256 + threadIdx.x;
  const float sd = stats[1], lo = stats[2], hi = stats[3];
  const float inv = 1.f / (sd + 1e-8f);
  const float v = x[i];
  const float s = (v - lo) * inv + 1e-8f;
  const float r = (hi - lo) * inv + 1e-8f;
  x[i] = s / r;
}

extern "C" void kernel_launch(void* const* d_in, const int* in_sizes, int n_in,
                              void* d_out, int out_size, void* d_ws, size_t ws_size,
                              hipStream_t stream) {
  const float* vol = (const float*)d_in[0];  // (1,1,256,256,256)
  const float* Rm  = (const float*)d_in[1];  // (1,3,3)
  const float* Tv  = (const float*)d_in[2];  // (1,3)
  const float* fo  = (const float*)d_in[3];  // (1,)
  float* out   = (float*)d_out;              // 65536 f32 (1,1,256,256)
  float* stats = (float*)d_ws;               // 4 floats

  render_kernel<<<IMG, IMG, 0, stream>>>(vol, Rm, Tv, fo, out);
  reduce_kernel<<<1, 256, 0, stream>>>(out, stats);
  norm_kernel<<<IMG, IMG, 0, stream>>>(out, stats);
}